// HGT_9302899163697
// MI455X (gfx1250) — compile-verified
//
#include <hip/hip_runtime.h>
#include <math.h>

// Problem constants (fixed by the reference)
#define NN  50000
#define DD  128
#define HH  8
#define HDD 16
#define LLAYERS 2
#define TT  4
#define RR  8
#define EE  800000

typedef float v2f __attribute__((ext_vector_type(2)));
typedef float v8f __attribute__((ext_vector_type(8)));

// ---------------------------------------------------------------------------
// Setup: rank-within-type (== cumsum(onehot)-1 in the reference), per-type
// counts/offsets, and a type-grouped permutation so GEMM tiles are
// type-uniform. Deterministic: each of the 4 threads serially scans its type.
// ---------------------------------------------------------------------------
__global__ void k_rank_perm(const int* __restrict__ ntype, int* __restrict__ rank,
                            int* __restrict__ perm, int* __restrict__ cnt_off) {
  __shared__ int scnt[TT];
  int t = threadIdx.x;
  if (t < TT) {
    int c = 0;
    for (int i = 0; i < NN; ++i)
      if (ntype[i] == t) rank[i] = c++;
    scnt[t] = c;
  }
  __syncthreads();
  if (t < TT) {
    int off = 0;
    for (int j = 0; j < t; ++j) off += scnt[j];
    cnt_off[t] = scnt[t];       // counts
    cnt_off[TT + t] = off;      // exclusive offsets
    int c = 0;
    for (int i = 0; i < NN; ++i)
      if (ntype[i] == t) perm[off + c++] = i;
  }
}

// h[i,:] = embed[ntype[i], rank[i], :]
__global__ void k_init_h(const float* __restrict__ embed, const int* __restrict__ ntype,
                         const int* __restrict__ rank, float* __restrict__ h) {
  int i = blockIdx.x * blockDim.x + threadIdx.x;
  if (i >= NN * DD) return;
  int n = i / DD, d = i % DD;
  h[i] = embed[((size_t)ntype[n] * NN + rank[n]) * DD + d];
}

// ---------------------------------------------------------------------------
// Fused typed k/q/v projection. Grid = (ceil(N/16), T). Block = 256 (8 waves).
// Each block: one 16-row tile of the permuted node list (type-uniform).
// Each wave: one 16-col slab. f32 WMMA 16x16x4, K-loop over D=128.
// A-frag layout (16x4 f32): lanes 0-15 rows, VGPR0/1 = K0/K1; lanes 16-31 K2/K3.
// ---------------------------------------------------------------------------
__global__ void __launch_bounds__(256)
k_gemm_kqv(const float* __restrict__ x,
           const float* __restrict__ Wk, const float* __restrict__ Wq,
           const float* __restrict__ Wv,
           const int* __restrict__ perm, const int* __restrict__ cnt_off,
           float* __restrict__ Kf, float* __restrict__ Qf, float* __restrict__ Vf) {
  int t = blockIdx.y;
  int cnt = cnt_off[t], off = cnt_off[TT + t];
  int row0 = (int)blockIdx.x * 16;
  if (row0 >= cnt) return;

  __shared__ float sx[16][DD];   // 8 KB A tile
  __shared__ int   snode[16];
  if (threadIdx.x < 16) {
    int rr = row0 + (int)threadIdx.x;
    snode[threadIdx.x] = (rr < cnt) ? perm[off + rr] : -1;
  }
  __syncthreads();
  for (int i = threadIdx.x; i < 16 * DD; i += 256) {
    int r = i / DD, c = i % DD;
    int n = snode[r];
    sx[r][c] = (n >= 0) ? x[(size_t)n * DD + c] : 0.0f;
  }
  __syncthreads();

  int wave = threadIdx.x >> 5;
  int lane = threadIdx.x & 31;
  int lr   = lane & 15;                 // A row / B col within tile
  int kh   = (lane < 16) ? 0 : 2;       // K half
  int colg = wave * 16 + lr;            // global output column

  const float* WkT = Wk + (size_t)t * DD * DD;
  const float* WqT = Wq + (size_t)t * DD * DD;
  const float* WvT = Wv + (size_t)t * DD * DD;

  v8f ck = {}, cq = {}, cv = {};
  for (int k0 = 0; k0 < DD; k0 += 4) {
    v2f a;
    a.x = sx[lr][k0 + kh];
    a.y = sx[lr][k0 + kh + 1];
    int r0 = (k0 + kh) * DD + colg;
    v2f bk, bq, bv;
    bk.x = WkT[r0]; bk.y = WkT[r0 + DD];
    bq.x = WqT[r0]; bq.y = WqT[r0 + DD];
    bv.x = WvT[r0]; bv.y = WvT[r0 + DD];
    ck = __builtin_amdgcn_wmma_f32_16x16x4_f32(false, a, false, bk, (short)0, ck, false, false);
    cq = __builtin_amdgcn_wmma_f32_16x16x4_f32(false, a, false, bq, (short)0, cq, false, false);
    cv = __builtin_amdgcn_wmma_f32_16x16x4_f32(false, a, false, bv, (short)0, cv, false, false);
  }

  int rb = (lane < 16) ? 0 : 8;
#pragma unroll
  for (int j = 0; j < 8; ++j) {
    int n = snode[rb + j];
    if (n >= 0) {
      size_t p = (size_t)n * DD + colg;
      Kf[p] = ck[j]; Qf[p] = cq[j]; Vf[p] = cv[j];
    }
  }
}

// ---------------------------------------------------------------------------
// Typed output projection Wa + fused sigmoid-skip blend (alpha uniform per
// type tile):  hc = (hagg @ Wa[t]) * alpha + x * (1-alpha)
// ---------------------------------------------------------------------------
__global__ void __launch_bounds__(256)
k_gemm_wa(const float* __restrict__ hagg, const float* __restrict__ Wa,
          const float* __restrict__ x, const float* __restrict__ skip_l,
          const int* __restrict__ perm, const int* __restrict__ cnt_off,
          float* __restrict__ hc) {
  int t = blockIdx.y;
  int cnt = cnt_off[t], off = cnt_off[TT + t];
  int row0 = (int)blockIdx.x * 16;
  if (row0 >= cnt) return;

  __shared__ float sx[16][DD];
  __shared__ int   snode[16];
  if (threadIdx.x < 16) {
    int rr = row0 + (int)threadIdx.x;
    snode[threadIdx.x] = (rr < cnt) ? perm[off + rr] : -1;
  }
  __syncthreads();
  for (int i = threadIdx.x; i < 16 * DD; i += 256) {
    int r = i / DD, c = i % DD;
    int n = snode[r];
    sx[r][c] = (n >= 0) ? hagg[(size_t)n * DD + c] : 0.0f;
  }
  __syncthreads();

  int wave = threadIdx.x >> 5;
  int lane = threadIdx.x & 31;
  int lr   = lane & 15;
  int kh   = (lane < 16) ? 0 : 2;
  int colg = wave * 16 + lr;
  const float* WT = Wa + (size_t)t * DD * DD;

  v8f cacc = {};
  for (int k0 = 0; k0 < DD; k0 += 4) {
    v2f a;
    a.x = sx[lr][k0 + kh];
    a.y = sx[lr][k0 + kh + 1];
    int r0 = (k0 + kh) * DD + colg;
    v2f b;
    b.x = WT[r0]; b.y = WT[r0 + DD];
    cacc = __builtin_amdgcn_wmma_f32_16x16x4_f32(false, a, false, b, (short)0, cacc, false, false);
  }

  float alpha = 1.0f / (1.0f + expf(-skip_l[t]));
  int rb = (lane < 16) ? 0 : 8;
#pragma unroll
  for (int j = 0; j < 8; ++j) {
    int n = snode[rb + j];
    if (n >= 0) {
      size_t p = (size_t)n * DD + colg;
      hc[p] = cacc[j] * alpha + x[p] * (1.0f - alpha);
    }
  }
}

// ---------------------------------------------------------------------------
// Edge pass A: a[e,h] = (k[src]^T * ratt[h,r] * q[dst]) * pri[h,r] / 4
// + order-invariant segment max via monotone-uint atomicMax.
// All 64 relation matrices (64 KB) staged in LDS.
// ---------------------------------------------------------------------------
__global__ void __launch_bounds__(256)
k_edge_logits(const float* __restrict__ Kf, const float* __restrict__ Qf,
              const float* __restrict__ ratt, const float* __restrict__ rpri,
              const int* __restrict__ src, const int* __restrict__ dst,
              const int* __restrict__ etype,
              float* __restrict__ a_out, unsigned int* __restrict__ amax_enc) {
  __shared__ float sM[HH * RR * HDD * HDD];   // 64 KB
  for (int i = threadIdx.x; i < HH * RR * HDD * HDD; i += 256) sM[i] = ratt[i];
  __syncthreads();

  int tid = blockIdx.x * blockDim.x + threadIdx.x;
  if (tid >= EE * HH) return;
  int e = tid >> 3, h = tid & 7;
  int s = src[e], d = dst[e], r = etype[e];

  const float4* kp = (const float4*)(Kf + (size_t)s * DD + h * HDD);
  const float4* qp = (const float4*)(Qf + (size_t)d * DD + h * HDD);
  float kr[HDD], qr[HDD];
#pragma unroll
  for (int i = 0; i < 4; ++i) {
    float4 a4 = kp[i];
    kr[4*i] = a4.x; kr[4*i+1] = a4.y; kr[4*i+2] = a4.z; kr[4*i+3] = a4.w;
    float4 b4 = qp[i];
    qr[4*i] = b4.x; qr[4*i+1] = b4.y; qr[4*i+2] = b4.z; qr[4*i+3] = b4.w;
  }
  const float* M = &sM[(h * RR + r) * HDD * HDD];
  float acc = 0.0f;
#pragma unroll
  for (int dd = 0; dd < HDD; ++dd) {
    float tq = 0.0f;
#pragma unroll
    for (int o = 0; o < HDD; ++o) tq = fmaf(M[dd * HDD + o], qr[o], tq);
    acc = fmaf(kr[dd], tq, acc);
  }
  acc *= rpri[h * RR + r] * 0.25f;   // / sqrt(16)
  a_out[tid] = acc;

  unsigned int b = __float_as_uint(acc);
  unsigned int key = (b & 0x80000000u) ? ~b : (b | 0x80000000u);
  atomicMax(&amax_enc[(size_t)d * HH + h], key);
}

// Decode monotone key -> float; non-finite (incl. "no incoming edges") -> 0.
__global__ void k_decode_max(const unsigned int* __restrict__ enc, float* __restrict__ amax) {
  int i = blockIdx.x * blockDim.x + threadIdx.x;
  if (i >= NN * HH) return;
  unsigned int key = enc[i];
  unsigned int b = (key & 0x80000000u) ? (key & 0x7FFFFFFFu) : ~key;
  amax[i] = ((b & 0x7F800000u) == 0x7F800000u) ? 0.0f : __uint_as_float(b);
}

// ex = exp(a - amax[dst]); asum[dst] += ex  (a buffer updated in place)
__global__ void k_exp_sum(const int* __restrict__ dst, const float* __restrict__ amax,
                          float* __restrict__ a_io, float* __restrict__ asum) {
  int tid = blockIdx.x * blockDim.x + threadIdx.x;
  if (tid >= EE * HH) return;
  int e = tid >> 3, h = tid & 7;
  int d = dst[e];
  float ex = expf(a_io[tid] - amax[(size_t)d * HH + h]);
  a_io[tid] = ex;
  atomicAdd(&asum[(size_t)d * HH + h], ex);
}

// hagg[dst,h,:] += (v[src,h,:] @ rmsg[h,r]) * ex/(asum[dst,h]+1e-9)
__global__ void __launch_bounds__(256)
k_aggregate(const float* __restrict__ Vf, const float* __restrict__ rmsg,
            const int* __restrict__ src, const int* __restrict__ dst,
            const int* __restrict__ etype,
            const float* __restrict__ ex, const float* __restrict__ asum,
            float* __restrict__ hagg) {
  __shared__ float sM[HH * RR * HDD * HDD];   // 64 KB
  for (int i = threadIdx.x; i < HH * RR * HDD * HDD; i += 256) sM[i] = rmsg[i];
  __syncthreads();

  int tid = blockIdx.x * blockDim.x + threadIdx.x;
  if (tid >= EE * HH) return;
  int e = tid >> 3, h = tid & 7;
  int s = src[e], d = dst[e], r = etype[e];

  float att = ex[tid] / (asum[(size_t)d * HH + h] + 1e-9f);
  const float4* vp = (const float4*)(Vf + (size_t)s * DD + h * HDD);
  float vr[HDD];
#pragma unroll
  for (int i = 0; i < 4; ++i) {
    float4 a4 = vp[i];
    vr[4*i] = a4.x; vr[4*i+1] = a4.y; vr[4*i+2] = a4.z; vr[4*i+3] = a4.w;
  }
  const float* M = &sM[(h * RR + r) * HDD * HDD];
  float* op = hagg + (size_t)d * DD + h * HDD;
#pragma unroll
  for (int o = 0; o < HDD; ++o) {
    float m = 0.0f;
#pragma unroll
    for (int dd = 0; dd < HDD; ++dd) m = fmaf(vr[dd], M[dd * HDD + o], m);
    atomicAdd(&op[o], m * att);
  }
}

// out = LayerNorm(hc) * g + b + x   (block = 128 threads = one row)
__global__ void __launch_bounds__(128)
k_ln_res(const float* __restrict__ hc, const float* __restrict__ x,
         const float* __restrict__ g, const float* __restrict__ b,
         float* __restrict__ out) {
  __shared__ float sred[DD];
  int row = blockIdx.x, d = threadIdx.x;
  float v = hc[(size_t)row * DD + d];
  sred[d] = v;
  __syncthreads();
  for (int s = 64; s > 0; s >>= 1) {
    if (d < s) sred[d] += sred[d + s];
    __syncthreads();
  }
  float mu = sred[0] * (1.0f / DD);
  __syncthreads();
  float diff = v - mu;
  sred[d] = diff * diff;
  __syncthreads();
  for (int s = 64; s > 0; s >>= 1) {
    if (d < s) sred[d] += sred[d + s];
    __syncthreads();
  }
  float var = sred[0] * (1.0f / DD);
  out[(size_t)row * DD + d] =
      diff * rsqrtf(var + 1e-5f) * g[d] + b[d] + x[(size_t)row * DD + d];
}

// ---------------------------------------------------------------------------
extern "C" void kernel_launch(void* const* d_in, const int* in_sizes, int n_in,
                              void* d_out, int out_size, void* d_ws, size_t ws_size,
                              hipStream_t stream) {
  (void)in_sizes; (void)n_in; (void)out_size; (void)ws_size;
  const float* embed = (const float*)d_in[0];
  const float* Wk    = (const float*)d_in[1];
  const float* Wq    = (const float*)d_in[2];
  const float* Wv    = (const float*)d_in[3];
  const float* Wa    = (const float*)d_in[4];
  const float* ratt  = (const float*)d_in[5];
  const float* rmsg  = (const float*)d_in[6];
  const float* rpri  = (const float*)d_in[7];
  const float* skip  = (const float*)d_in[8];
  const float* lng   = (const float*)d_in[9];
  const float* lnb   = (const float*)d_in[10];
  const int*   src   = (const int*)d_in[11];
  const int*   dst   = (const int*)d_in[12];
  const int*   ntype = (const int*)d_in[13];
  const int*   etype = (const int*)d_in[14];
  float* out = (float*)d_out;

  const size_t nd = (size_t)NN * DD;
  char* w = (char*)d_ws;
  float* bufA = (float*)w; w += nd * 4;
  float* bufB = (float*)w; w += nd * 4;
  float* Kf   = (float*)w; w += nd * 4;      // reused as hc after edge passes
  float* Qf   = (float*)w; w += nd * 4;
  float* Vf   = (float*)w; w += nd * 4;
  float* hagg = (float*)w; w += nd * 4;
  float* aex  = (float*)w; w += (size_t)EE * HH * 4;
  unsigned int* amax_enc = (unsigned int*)w; w += (size_t)NN * HH * 4;
  float* amaxf = (float*)w; w += (size_t)NN * HH * 4;
  float* asum  = (float*)w; w += (size_t)NN * HH * 4;
  int* rank    = (int*)w;   w += (size_t)NN * 4;
  int* perm    = (int*)w;   w += (size_t)NN * 4;
  int* cnt_off = (int*)w;   w += 2 * TT * 4;
  float* hc = Kf;  // alias: Kf dead before Wa GEMM within each layer

  k_rank_perm<<<1, 32, 0, stream>>>(ntype, rank, perm, cnt_off);
  k_init_h<<<(NN * DD + 255) / 256, 256, 0, stream>>>(embed, ntype, rank, bufA);

  dim3 ggrid((NN + 15) / 16, TT);
  int egrid = (EE * HH + 255) / 256;

  const float* x = bufA;
  for (int l = 0; l < LLAYERS; ++l) {
    float* y = (l == LLAYERS - 1) ? out : bufB;
    const float* Wk_l   = Wk   + (size_t)l * TT * DD * DD;
    const float* Wq_l   = Wq   + (size_t)l * TT * DD * DD;
    const float* Wv_l   = Wv   + (size_t)l * TT * DD * DD;
    const float* Wa_l   = Wa   + (size_t)l * TT * DD * DD;
    const float* ratt_l = ratt + (size_t)l * HH * RR * HDD * HDD;
    const float* rmsg_l = rmsg + (size_t)l * HH * RR * HDD * HDD;
    const float* rpri_l = rpri + (size_t)l * HH * RR;
    const float* skip_l = skip + (size_t)l * TT;
    const float* lng_l  = lng  + (size_t)l * DD;
    const float* lnb_l  = lnb  + (size_t)l * DD;

    hipMemsetAsync(amax_enc, 0, (size_t)NN * HH * 4, stream);  // key 0 == -NaN -> decodes to 0
    hipMemsetAsync(asum,     0, (size_t)NN * HH * 4, stream);
    hipMemsetAsync(hagg,     0, nd * 4, stream);

    k_gemm_kqv<<<ggrid, 256, 0, stream>>>(x, Wk_l, Wq_l, Wv_l, perm, cnt_off, Kf, Qf, Vf);
    k_edge_logits<<<egrid, 256, 0, stream>>>(Kf, Qf, ratt_l, rpri_l, src, dst, etype, aex, amax_enc);
    k_decode_max<<<(NN * HH + 255) / 256, 256, 0, stream>>>(amax_enc, amaxf);
    k_exp_sum<<<egrid, 256, 0, stream>>>(dst, amaxf, aex, asum);
    k_aggregate<<<egrid, 256, 0, stream>>>(Vf, rmsg_l, src, dst, etype, aex, asum, hagg);
    k_gemm_wa<<<ggrid, 256, 0, stream>>>(hagg, Wa_l, x, skip_l, perm, cnt_off, hc);
    k_ln_res<<<NN, DD, 0, stream>>>(hc, x, lng_l, lnb_l, y);
    x = y;
  }
}